// TestModule_77601469104783
// MI455X (gfx1250) — compile-verified
//
#include <hip/hip_runtime.h>
#include <cstdint>

#define D_FEAT 64

// ---------------------------------------------------------------------------
// Kernel 1: zero the output accumulator (harness poisons d_out with 0xAA).
// 128-bit stores, fully coalesced.
// ---------------------------------------------------------------------------
__global__ void gnn_zero_out_kernel(float4* __restrict__ out, int n4) {
    int i = blockIdx.x * blockDim.x + threadIdx.x;
    if (i < n4) {
        out[i] = make_float4(0.f, 0.f, 0.f, 0.f);
    }
}

// ---------------------------------------------------------------------------
// Kernel 2: edge-parallel gather + scatter-add.
//   16 lanes cooperate on one edge: lane q handles floats [4q, 4q+4) of the
//   64-float feature row. Gather is one global_load_b128 per lane (256 B
//   contiguous per 16-lane group). Scatter is 4 non-returning hardware
//   global_atomic_add_f32 sharing one 64-bit address VGPR pair via the
//   offset: immediate. x (25.6 MB) and out (25.6 MB) both live in the
//   192 MB L2, so gathers and atomics resolve at L2 speed.
// ---------------------------------------------------------------------------
__device__ __forceinline__ void quad_atomic_add_f32(float* p, float4 v) {
    unsigned long long addr = (unsigned long long)(uintptr_t)p;
    asm volatile(
        "global_atomic_add_f32 %0, %1, off scope:SCOPE_DEV\n\t"
        "global_atomic_add_f32 %0, %2, off offset:4 scope:SCOPE_DEV\n\t"
        "global_atomic_add_f32 %0, %3, off offset:8 scope:SCOPE_DEV\n\t"
        "global_atomic_add_f32 %0, %4, off offset:12 scope:SCOPE_DEV"
        :
        : "v"(addr), "v"(v.x), "v"(v.y), "v"(v.z), "v"(v.w)
        : "memory");
}

__global__ void gnn_scatter_sum_kernel(const float* __restrict__ x,
                                       const int* __restrict__ row,
                                       const int* __restrict__ col,
                                       float* __restrict__ out,
                                       int n_edges) {
    const int tid = blockIdx.x * blockDim.x + threadIdx.x;
    const int e = tid >> 4;   // 16 lanes per edge
    const int q = tid & 15;   // which float4 of the 64-float row
    if (e >= n_edges) return;

    // Indices are streamed exactly once -> non-temporal (TH=NT) so they
    // don't displace the L2-resident x/out working set.
    const int r = __builtin_nontemporal_load(row + e);
    const int c = __builtin_nontemporal_load(col + e);

    // Coalesced 128-bit gather of this edge's feature quad.
    const float4 v = *(const float4*)(x + (size_t)r * D_FEAT + (size_t)q * 4);

    // Scatter-accumulate with hardware f32 atomics (device scope).
    float* dst = out + (size_t)c * D_FEAT + (size_t)q * 4;
    quad_atomic_add_f32(dst, v);
}

// ---------------------------------------------------------------------------
// Launch
// ---------------------------------------------------------------------------
extern "C" void kernel_launch(void* const* d_in, const int* in_sizes, int n_in,
                              void* d_out, int out_size, void* d_ws, size_t ws_size,
                              hipStream_t stream) {
    const float* x  = (const float*)d_in[0];       // [N, 64] fp32
    const int*   ei = (const int*)d_in[1];         // [2, E] int32 (JAX x64 off)
    float*       out = (float*)d_out;              // [N, 64] fp32

    const int n_edges = in_sizes[1] / 2;
    const int* row = ei;                            // ei[0, :]
    const int* col = ei + n_edges;                  // ei[1, :]

    // 1) zero the accumulator
    const int n4 = out_size / 4;
    const int zblocks = (n4 + 255) / 256;
    gnn_zero_out_kernel<<<zblocks, 256, 0, stream>>>((float4*)out, n4);

    // 2) scatter-sum: 16 work-items per edge
    const long long total = (long long)n_edges * 16;
    const int sblocks = (int)((total + 255) / 256);
    gnn_scatter_sum_kernel<<<sblocks, 256, 0, stream>>>(x, row, col, out, n_edges);
}